// BernNet_5342939316735
// MI455X (gfx1250) — compile-verified
//
#include <hip/hip_runtime.h>
#include <hip/hip_bf16.h>

typedef float v2f __attribute__((ext_vector_type(2)));
typedef float v8f __attribute__((ext_vector_type(8)));

#define IN_F 256
#define HID  128
#define OUT_F 64
#define KCH  10
#define NLAYERS 2

// ---------------- CDNA5 async global->LDS helpers ----------------
// ISA 15.18.3: GLOBAL_LOAD_ASYNC_TO_LDS_B128 — VDST holds LDS byte address,
// VADDR holds 64-bit global address; tracked with ASYNCcnt.
// Generic pointer to __shared__ truncated to 32 bits == wave-relative LDS addr
// (flat aperture mapping: the low 32 address bits are the LDS address).

__device__ __forceinline__ void async_ld16(void* lds_ptr, const void* gptr) {
    unsigned l32 = (unsigned)(uintptr_t)lds_ptr;
    asm volatile("global_load_async_to_lds_b128 %0, %1, off"
                 :: "v"(l32), "v"(gptr) : "memory");
}

__device__ __forceinline__ void wait_async_all() {
    asm volatile("s_wait_asynccnt 0x0" ::: "memory");
}

// ---------------- elementwise helpers ----------------

__global__ void fill_zero_kernel(float* __restrict__ p, long n) {
    long i = (long)blockIdx.x * blockDim.x + threadIdx.x;
    if (i < n) p[i] = 0.0f;
}

__global__ void deg_kernel(const int* __restrict__ src, float* __restrict__ deg, long E) {
    long e = (long)blockIdx.x * blockDim.x + threadIdx.x;
    if (e < E) atomicAdd(&deg[src[e]], 1.0f);
}

__global__ void dinv_kernel(const float* __restrict__ deg, float* __restrict__ dinv, long N) {
    long i = (long)blockIdx.x * blockDim.x + threadIdx.x;
    if (i < N) {
        float d = deg[i];
        dinv[i] = (d > 0.0f) ? rsqrtf(fmaxf(d, 1e-30f)) : 0.0f;
    }
}

__global__ void wedge_kernel(const int* __restrict__ src, const int* __restrict__ dst,
                             const float* __restrict__ dinv, float* __restrict__ w, long E) {
    long e = (long)blockIdx.x * blockDim.x + threadIdx.x;
    if (e < E) w[e] = -0.5f * dinv[src[e]] * dinv[dst[e]];
}

// ---------------- SpMM scatter: agg[dst] += w * x[src]  (one wave / edge) ----------------

__global__ __launch_bounds__(256) void spmm_scatter_kernel(
    const float* __restrict__ x, const int* __restrict__ src, const int* __restrict__ dst,
    const float* __restrict__ w, float* __restrict__ agg, long E) {
    long wid = (long)blockIdx.x * (blockDim.x >> 5) + (threadIdx.x >> 5);
    int lane = threadIdx.x & 31;
    if (wid >= E) return;
    int   s  = src[wid];           // uniform across wave -> scalarized
    int   d  = dst[wid];
    float we = w[wid];
    const float4 v = ((const float4*)(x + (long)s * HID))[lane];  // 32 lanes x 4 = 128 feats
    float* out = agg + (long)d * HID + lane * 4;
    atomicAdd(out + 0, we * v.x);
    atomicAdd(out + 1, we * v.y);
    atomicAdd(out + 2, we * v.z);
    atomicAdd(out + 3, we * v.w);
}

// ---------------- Chebyshev recurrence (float4) ----------------

// tx1 = 0.5*h - agg ; out = c0*h + c1*tx1
__global__ void cheb_first_kernel(const float4* __restrict__ h, const float4* __restrict__ agg,
                                  const float* __restrict__ coeffs, int ci,
                                  float4* __restrict__ tx1, float4* __restrict__ out, long n4) {
    long i = (long)blockIdx.x * blockDim.x + threadIdx.x;
    if (i >= n4) return;
    float c0 = coeffs[ci], c1 = coeffs[ci + 1];
    float4 hv = h[i], av = agg[i], t, o;
    t.x = 0.5f * hv.x - av.x;  t.y = 0.5f * hv.y - av.y;
    t.z = 0.5f * hv.z - av.z;  t.w = 0.5f * hv.w - av.w;
    o.x = c0 * hv.x + c1 * t.x;  o.y = c0 * hv.y + c1 * t.y;
    o.z = c0 * hv.z + c1 * t.z;  o.w = c0 * hv.w + c1 * t.w;
    tx1[i] = t;  out[i] = o;
}

// tx2 = tx1 - 2*agg - tx0 ; out += ck*tx2
__global__ void cheb_step_kernel(const float4* __restrict__ tx1, const float4* __restrict__ agg,
                                 const float4* __restrict__ tx0, const float* __restrict__ coeffs,
                                 int ci, float4* __restrict__ tx2, float4* __restrict__ out, long n4) {
    long i = (long)blockIdx.x * blockDim.x + threadIdx.x;
    if (i >= n4) return;
    float ck = coeffs[ci];
    float4 t1 = tx1[i], av = agg[i], t0 = tx0[i], o = out[i], t2;
    t2.x = t1.x - 2.0f * av.x - t0.x;  t2.y = t1.y - 2.0f * av.y - t0.y;
    t2.z = t1.z - 2.0f * av.z - t0.z;  t2.w = t1.w - 2.0f * av.w - t0.w;
    o.x += ck * t2.x;  o.y += ck * t2.y;  o.z += ck * t2.z;  o.w += ck * t2.w;
    tx2[i] = t2;  out[i] = o;
}

__global__ void relu_copy_kernel(const float4* __restrict__ in, float4* __restrict__ out, long n4) {
    long i = (long)blockIdx.x * blockDim.x + threadIdx.x;
    if (i >= n4) return;
    float4 v = in[i];
    v.x = fmaxf(v.x, 0.0f); v.y = fmaxf(v.y, 0.0f);
    v.z = fmaxf(v.z, 0.0f); v.w = fmaxf(v.w, 0.0f);
    out[i] = v;
}

// ---------------- WMMA fp32 GEMMs (V_WMMA_F32_16X16X4_F32) ----------------
// One wave computes one 16x16 output tile. A-frag (16x4 f32, 2 VGPR):
// lanes 0-15 hold K=0,1 for M=lane; lanes 16-31 hold K=2,3 for M=lane-16.
// B-frag (4x16): lanes 0-15 hold N=lane for K=0,1; lanes 16-31 N=lane-16 for K=2,3.
// C/D: VGPR r -> M=r (lanes 0-15) / M=r+8 (lanes 16-31), N=lane&15.
// A-tile (16 rows, contiguous in row-major X) is staged once per block into LDS
// via GLOBAL_LOAD_ASYNC_TO_LDS_B128 and shared by all column-tile waves.

// H[M x HID] = relu(X[M x IN_F] @ W1[IN_F x HID] + b1)  — 8 waves/block = 8 col tiles
__global__ __launch_bounds__(256) void gemm1_relu_wmma(
    const float* __restrict__ X, const float* __restrict__ W, const float* __restrict__ bias,
    float* __restrict__ H, int M) {
    __shared__ float xt[16 * IN_F];                 // 16 KB A-tile
    int tid  = threadIdx.x;
    int row0 = blockIdx.x * 16;

    // stage contiguous 16KB tile: 1024 x b128, 256 threads x 4 iterations
    const char* gsrc = (const char*)(X + (long)row0 * IN_F);
    #pragma unroll
    for (int it = 0; it < 4; ++it) {
        int boff = (it * 256 + tid) * 16;
        async_ld16((char*)xt + boff, gsrc + boff);
    }
    wait_async_all();
    __syncthreads();

    int wave = tid >> 5;               // column tile 0..7
    int lane = tid & 31;
    int col0 = wave * 16;
    int mn   = lane & 15;              // M index for A, N index for B
    int kb   = (lane >> 4) << 1;       // 0 or 2
    const float* arow = xt + mn * IN_F;
    v8f acc = {};
    for (int k = 0; k < IN_F; k += 4) {
        v2f a, b;
        float2 av = *(const float2*)(arow + k + kb);   // ds_load_b64
        a.x = av.x;  a.y = av.y;
        b.x = W[(long)(k + kb)     * HID + col0 + mn];
        b.y = W[(long)(k + kb + 1) * HID + col0 + mn];
        acc = __builtin_amdgcn_wmma_f32_16x16x4_f32(false, a, false, b,
                                                    (short)0, acc, false, false);
    }
    float bcol = bias[col0 + mn];
    int rowadd = (lane >> 4) << 3;     // +8 for lanes 16-31
    #pragma unroll
    for (int r = 0; r < 8; ++r) {
        int row = row0 + r + rowadd;
        if (row < M) {
            float v = acc[r] + bcol;
            H[(long)row * HID + col0 + mn] = v > 0.0f ? v : 0.0f;
        }
    }
}

// OUT[M x OUT_F] = H[M x HID] @ W2[HID x OUT_F] + b2 — 4 waves/block = 4 col tiles
__global__ __launch_bounds__(128) void gemm2_wmma(
    const float* __restrict__ Hm, const float* __restrict__ W, const float* __restrict__ bias,
    float* __restrict__ O, int M) {
    __shared__ float ht[16 * HID];                  // 8 KB A-tile
    int tid  = threadIdx.x;
    int row0 = blockIdx.x * 16;

    const char* gsrc = (const char*)(Hm + (long)row0 * HID);
    #pragma unroll
    for (int it = 0; it < 4; ++it) {
        int boff = (it * 128 + tid) * 16;
        async_ld16((char*)ht + boff, gsrc + boff);
    }
    wait_async_all();
    __syncthreads();

    int wave = tid >> 5;               // column tile 0..3
    int lane = tid & 31;
    int col0 = wave * 16;
    int mn   = lane & 15;
    int kb   = (lane >> 4) << 1;
    const float* arow = ht + mn * HID;
    v8f acc = {};
    for (int k = 0; k < HID; k += 4) {
        v2f a, b;
        float2 av = *(const float2*)(arow + k + kb);   // ds_load_b64
        a.x = av.x;  a.y = av.y;
        b.x = W[(long)(k + kb)     * OUT_F + col0 + mn];
        b.y = W[(long)(k + kb + 1) * OUT_F + col0 + mn];
        acc = __builtin_amdgcn_wmma_f32_16x16x4_f32(false, a, false, b,
                                                    (short)0, acc, false, false);
    }
    float bcol = bias[col0 + mn];
    int rowadd = (lane >> 4) << 3;
    #pragma unroll
    for (int r = 0; r < 8; ++r) {
        int row = row0 + r + rowadd;
        if (row < M) O[(long)row * OUT_F + col0 + mn] = acc[r] + bcol;
    }
}

// ---------------- orchestration ----------------

static inline int nblk(long n, int b) { return (int)((n + b - 1) / b); }

extern "C" void kernel_launch(void* const* d_in, const int* in_sizes, int n_in,
                              void* d_out, int out_size, void* d_ws, size_t ws_size,
                              hipStream_t stream) {
    const float* x      = (const float*)d_in[0];
    const int*   ei     = (const int*)  d_in[1];   // JAX default x64-off -> int32
    const float* W1     = (const float*)d_in[2];
    const float* b1     = (const float*)d_in[3];
    const float* coeffs = (const float*)d_in[4];   // (NLAYERS, K+1)
    const float* W2     = (const float*)d_in[5];
    const float* b2     = (const float*)d_in[6];

    const long N = (long)in_sizes[0] / IN_F;
    const long E = (long)in_sizes[1] / 2;
    const int* src = ei;
    const int* dst = ei + E;

    const long NH = N * HID;
    char* ws = (char*)d_ws;
    float* h    = (float*)ws;  ws += NH * sizeof(float);
    float* bufB = (float*)ws;  ws += NH * sizeof(float);
    float* bufC = (float*)ws;  ws += NH * sizeof(float);
    float* agg  = (float*)ws;  ws += NH * sizeof(float);
    float* outb = (float*)ws;  ws += NH * sizeof(float);
    float* deg  = (float*)ws;  ws += N  * sizeof(float);
    float* dinv = (float*)ws;  ws += N  * sizeof(float);
    float* wedg = (float*)ws;  ws += E  * sizeof(float);

    const long n4 = NH / 4;
    const int  TB = 256;

    // ---- graph normalization ----
    fill_zero_kernel<<<nblk(N, TB), TB, 0, stream>>>(deg, N);
    deg_kernel      <<<nblk(E, TB), TB, 0, stream>>>(src, deg, E);
    dinv_kernel     <<<nblk(N, TB), TB, 0, stream>>>(deg, dinv, N);
    wedge_kernel    <<<nblk(E, TB), TB, 0, stream>>>(src, dst, dinv, wedg, E);

    // ---- h = relu(x @ W1 + b1) ----
    gemm1_relu_wmma<<<nblk(N, 16), 256, 0, stream>>>(x, W1, b1, h, (int)N);

    // ---- 2x BernConv layers ----
    const int sblocks = (int)((E + 7) / 8);   // 8 waves (edges) per block

    for (int l = 0; l < NLAYERS; ++l) {
        const int ci = l * (KCH + 1);

        fill_zero_kernel   <<<nblk(NH, TB), TB, 0, stream>>>(agg, NH);
        spmm_scatter_kernel<<<sblocks, 256, 0, stream>>>(h, src, dst, wedg, agg, E);
        cheb_first_kernel  <<<nblk(n4, TB), TB, 0, stream>>>(
            (const float4*)h, (const float4*)agg, coeffs, ci,
            (float4*)bufB, (float4*)outb, n4);

        float* t0 = h;  float* t1 = bufB;  float* t2 = bufC;
        for (int k = 2; k <= KCH; ++k) {
            fill_zero_kernel   <<<nblk(NH, TB), TB, 0, stream>>>(agg, NH);
            spmm_scatter_kernel<<<sblocks, 256, 0, stream>>>(t1, src, dst, wedg, agg, E);
            cheb_step_kernel   <<<nblk(n4, TB), TB, 0, stream>>>(
                (const float4*)t1, (const float4*)agg, (const float4*)t0,
                coeffs, ci + k, (float4*)t2, (float4*)outb, n4);
            float* tmp = t0; t0 = t1; t1 = t2; t2 = tmp;   // rotate
        }
        // h_next = relu(out); t2 holds the stalest (free) buffer
        relu_copy_kernel<<<nblk(n4, TB), TB, 0, stream>>>(
            (const float4*)outb, (float4*)t2, n4);
        h = t2;  bufB = t0;  bufC = t1;
    }

    // ---- d_out = h @ W2 + b2 ----
    gemm2_wmma<<<nblk(N, 16), 128, 0, stream>>>(h, W2, b2, (float*)d_out, (int)N);
}